// Global_Filter_21363167330677
// MI455X (gfx1250) — compile-verified
//
#include <hip/hip_runtime.h>
#include <hip/hip_bf16.h>

// Global filter = irfftn(rfftn(x) * W): two small dense GEMMs (F: 540x450,
// G: 450x540) with a per-(freq,channel) complex diagonal in between.
// f32 WMMA (16x16x4), intermediate kept in LDS, F/G built once into d_ws.

typedef float v2f __attribute__((ext_vector_type(2)));
typedef float v8f __attribute__((ext_vector_type(8)));

#define K1PAIRS 226   // K=452 (450 padded to mult of 4) -> 226 pairs
#define M1PAD   544   // 540 freq rows padded to 34*16
#define K2PAIRS 270   // K=540 -> 270 pairs
#define M2PAD   480   // 450 spatial rows padded to 30*16 (for 2x2 blocking)
#define FP_ELEMS (K1PAIRS * M1PAD * 2)   // 245888 floats
#define GP_ELEMS (K2PAIRS * M2PAD * 2)   // 259200 floats

static __device__ __forceinline__ v8f wmma4(v2f a, v2f b, v8f c) {
  return __builtin_amdgcn_wmma_f32_16x16x4_f32(
      /*neg_a=*/false, a, /*neg_b=*/false, b,
      /*c_mod=*/(short)0, c, /*reuse_a=*/false, /*reuse_b=*/false);
}

// Build F (forward rfftn, ortho) and G (inverse w/ Hermitian weights) into ws.
// Layout: Fp[kpair*1088 + m*2 + j] = F[m][2*kpair+j]; same pattern for Gp.
__global__ void gf_build(float* __restrict__ Fp, float* __restrict__ Gp) {
  const float S = 0.04714045207910317f;     // 1/sqrt(450)
  const float TWO_PI = 6.283185307179586f;
  const int total1 = FP_ELEMS, total2 = GP_ELEMS;
  for (int i = blockIdx.x * blockDim.x + threadIdx.x; i < total1 + total2;
       i += gridDim.x * blockDim.x) {
    if (i < total1) {
      int kp = i / (M1PAD * 2), rem = i % (M1PAD * 2);
      int m = rem >> 1, j = rem & 1;
      int K = 2 * kp + j;                    // spatial index n = h*50+w*5+d
      float v = 0.f;
      if (m < 540 && K < 450) {
        int k = m >> 1, ri = m & 1;          // freq k = kh*30+kw*3+kd
        int kh = k / 30, kw = (k / 3) % 10, kd = k % 3;
        int h = K / 50, w = (K / 5) % 10, d = K % 5;
        float phi = TWO_PI * (kh * h / 9.f + kw * w / 10.f + kd * d / 5.f);
        v = ri ? -S * sinf(phi) : S * cosf(phi);
      }
      Fp[i] = v;
    } else {
      int ii = i - total1;
      int kp = ii / (M2PAD * 2), rem = ii % (M2PAD * 2);
      int m = rem >> 1, j = rem & 1;
      int col = 2 * kp + j;                  // Y row index = 2*kfreq + ri
      float v = 0.f;
      if (m < 450) {
        int k = col >> 1, ri = col & 1;
        int kh = k / 30, kw = (k / 3) % 10, kd = k % 3;
        int h = m / 50, w = (m / 5) % 10, d = m % 5;
        float mk = (kd == 0) ? 1.f : 2.f;    // Hermitian doubling, kd=1,2
        float phi = TWO_PI * (kh * h / 9.f + kw * w / 10.f + kd * d / 5.f);
        v = ri ? -S * mk * sinf(phi) : S * mk * cosf(phi);
      }
      Gp[ii] = v;
    }
  }
}

__global__ __launch_bounds__(256) void gf_main(
    const float* __restrict__ x, const float* __restrict__ cw,
    const float* __restrict__ Fp, const float* __restrict__ Gp,
    float* __restrict__ out) {
  // LDS: x-tile (452x64) and Y (540x64), both K-pair interleaved [kp][c][2].
  __shared__ float Xsp[K1PAIRS * 128];   // 115.7 KB
  __shared__ float Ysp[K2PAIRS * 128];   // 138.2 KB
  const int c0 = blockIdx.x * 64;        // channel tile (16 tiles, last partial)
  const int b  = blockIdx.y;             // batch
  const int tid = threadIdx.x;
  const int lane = tid & 31, wave = tid >> 5;
  const int lm = lane & 15, hi = lane >> 4;   // WMMA half-wave select

  // ---- Stage 1: load x[b, :, c0:c0+64] into LDS (zero-padded) ----
  const float* xb = x + (size_t)b * 450000 + c0;
  for (int i = tid; i < 452 * 64; i += 256) {
    int n = i >> 6, c = i & 63;
    float v = 0.f;
    if (n < 450 && (c0 + c) < 1000) v = xb[n * 1000 + c];
    Xsp[(n >> 1) * 128 + c * 2 + (n & 1)] = v;
  }
  __syncthreads();

  // ---- Stage 2: Y = F * X  (M=544, N=64, K=452), 2x2 tile blocking ----
  for (int blk = wave; blk < 34; blk += 8) {
    int m0 = (blk >> 1) * 32, n0 = (blk & 1) * 32;
    v8f a00 = {}, a01 = {}, a10 = {}, a11 = {};
    for (int s = 0; s < 113; ++s) {
      int kp = 2 * s + hi;                        // lane's K-pair
      const float* pa = Fp + (size_t)kp * (M1PAD * 2) + (m0 + lm) * 2;
      v2f A0 = *(const v2f*)pa;
      v2f A1 = *(const v2f*)(pa + 32);
      const float* pb = Xsp + kp * 128 + (n0 + lm) * 2;
      v2f B0 = *(const v2f*)pb;
      v2f B1 = *(const v2f*)(pb + 32);
      a00 = wmma4(A0, B0, a00);
      a01 = wmma4(A0, B1, a01);
      a10 = wmma4(A1, B0, a10);
      a11 = wmma4(A1, B1, a11);
    }
    for (int r = 0; r < 8; ++r) {                 // D: VGPR r -> M=m0+r(+8 hi)
      int mA = m0 + r + hi * 8, nA = n0 + lm;
      if (mA < 540) {
        Ysp[(mA >> 1) * 128 + nA * 2 + (mA & 1)]        = a00[r];
        Ysp[(mA >> 1) * 128 + (nA + 16) * 2 + (mA & 1)] = a01[r];
      }
      int mB = mA + 16;
      if (mB < 540) {
        Ysp[(mB >> 1) * 128 + nA * 2 + (mB & 1)]        = a10[r];
        Ysp[(mB >> 1) * 128 + (nA + 16) * 2 + (mB & 1)] = a11[r];
      }
    }
  }
  __syncthreads();

  // ---- Stage 3: complex diagonal: (re,im) *= (wr,wi); adjacent in LDS ----
  for (int i = tid; i < 270 * 64; i += 256) {
    int k = i >> 6, c = i & 63;
    int cg = c0 + c;
    float wr = 0.f, wi = 0.f;
    if (cg < 1000) {
      wr = cw[(k * 1000 + cg) * 2 + 0];
      wi = cw[(k * 1000 + cg) * 2 + 1];
    }
    float re = Ysp[k * 128 + c * 2 + 0];
    float im = Ysp[k * 128 + c * 2 + 1];
    Ysp[k * 128 + c * 2 + 0] = re * wr - im * wi;
    Ysp[k * 128 + c * 2 + 1] = re * wi + im * wr;
  }
  __syncthreads();

  // ---- Stage 4: out = G * Y  (M=480, N=64, K=540), store from accums ----
  float* ob = out + (size_t)b * 450000 + c0;
  for (int blk = wave; blk < 30; blk += 8) {
    int m0 = (blk >> 1) * 32, n0 = (blk & 1) * 32;
    v8f a00 = {}, a01 = {}, a10 = {}, a11 = {};
    for (int s = 0; s < 135; ++s) {
      int kp = 2 * s + hi;
      const float* pa = Gp + (size_t)kp * (M2PAD * 2) + (m0 + lm) * 2;
      v2f A0 = *(const v2f*)pa;
      v2f A1 = *(const v2f*)(pa + 32);
      const float* pb = Ysp + kp * 128 + (n0 + lm) * 2;
      v2f B0 = *(const v2f*)pb;
      v2f B1 = *(const v2f*)(pb + 32);
      a00 = wmma4(A0, B0, a00);
      a01 = wmma4(A0, B1, a01);
      a10 = wmma4(A1, B0, a10);
      a11 = wmma4(A1, B1, a11);
    }
    for (int r = 0; r < 8; ++r) {
      int mA = m0 + r + hi * 8, nA = n0 + lm;
      if (mA < 450 && (c0 + nA)      < 1000) ob[mA * 1000 + nA]      = a00[r];
      if (mA < 450 && (c0 + nA + 16) < 1000) ob[mA * 1000 + nA + 16] = a01[r];
      int mB = mA + 16;
      if (mB < 450 && (c0 + nA)      < 1000) ob[mB * 1000 + nA]      = a10[r];
      if (mB < 450 && (c0 + nA + 16) < 1000) ob[mB * 1000 + nA + 16] = a11[r];
    }
  }
}

extern "C" void kernel_launch(void* const* d_in, const int* in_sizes, int n_in,
                              void* d_out, int out_size, void* d_ws, size_t ws_size,
                              hipStream_t stream) {
  const float* x  = (const float*)d_in[0];   // (256, 450, 1000) f32
  const float* cw = (const float*)d_in[1];   // (9, 10, 3, 1000, 2) f32
  float* Fp = (float*)d_ws;                  // 245888 floats
  float* Gp = Fp + FP_ELEMS;                 // 259200 floats (~2 MB total)

  gf_build<<<256, 256, 0, stream>>>(Fp, Gp);

  dim3 grid(16, 256);                        // 16 channel tiles x 256 batches
  gf_main<<<grid, 256, 0, stream>>>(x, cw, Fp, Gp, (float*)d_out);
}